// VNNGP_30958124270280
// MI455X (gfx1250) — compile-verified
//
#include <hip/hip_runtime.h>
#include <math.h>
#include <stdint.h>

#define M_IND 1024
#define N_PTS 65536
#define K_NN  16
#define JITTER 0.0001f
#define NWAVES 8

typedef __attribute__((ext_vector_type(2))) float v2f;
typedef __attribute__((ext_vector_type(8))) float v8f;

// ---------------------------------------------------------------------------
// Kernel 1: build Lu = tril(Lu_raw,-1) + diag(exp(diag)) and
//           Kzz_j = exp(-0.5*|Zi-Zj|^2) + JITTER*I   (== L@L.T of the ref)
// ---------------------------------------------------------------------------
__global__ __launch_bounds__(256) void prep_kernel(
    const float* __restrict__ Lu_raw, const float* __restrict__ Z,
    float* __restrict__ Lu, float* __restrict__ Kzz) {
  const int t = blockIdx.x * 256 + threadIdx.x;   // 0 .. 1024*1024-1
  const int i = t >> 10;
  const int j = t & 1023;

  const float lr = Lu_raw[t];
  Lu[t] = (i > j) ? lr : ((i == j) ? expf(lr) : 0.0f);

  const float dx = Z[i * 3 + 0] - Z[j * 3 + 0];
  const float dy = Z[i * 3 + 1] - Z[j * 3 + 1];
  const float dz = Z[i * 3 + 2] - Z[j * 3 + 2];
  float sq = fmaxf(dx * dx + dy * dy + dz * dz, 0.0f);
  Kzz[t] = expf(-0.5f * sq) + ((i == j) ? JITTER : 0.0f);
}

// ---------------------------------------------------------------------------
// Kernel 2: Su = Lu @ Lu^T with V_WMMA_F32_16X16X4_F32.
// Each wave computes a 16x64 output strip: one A fragment feeds 4 WMMAs.
// A-frag (16x4 f32, ISA 7.12.2): lanes 0-15 hold K=0(v0),K=1(v1); lanes 16-31 K=2,3.
// B-frag (4x16): v0 = row K=(lane>=16), v1 = row K=2+(lane>=16), col N = lane&15.
// C/D (16x16 f32): vgpr r -> M = r + 8*(lane>=16), N = lane&15.
// ---------------------------------------------------------------------------
__global__ __launch_bounds__(256) void su_gemm_wmma(
    const float* __restrict__ Lu, float* __restrict__ Su) {
  const int lane = threadIdx.x & 31;
  const int wave = threadIdx.x >> 5;
  const int tileM = blockIdx.y * NWAVES + wave;  // 0..63
  const int i0 = tileM * 16;
  const int j0 = blockIdx.x * 64;                // 4 N-tiles per wave
  const int m  = lane & 15;
  const int hi = lane >> 4;                      // 0 or 1

  v8f acc0 = {}, acc1 = {}, acc2 = {}, acc3 = {};
  const float* __restrict__ arow = Lu + (i0 + m) * M_IND + hi * 2;
  const float* __restrict__ b0 = Lu + (j0 + 0 * 16 + m) * M_IND;
  const float* __restrict__ b1 = Lu + (j0 + 1 * 16 + m) * M_IND;
  const float* __restrict__ b2 = Lu + (j0 + 2 * 16 + m) * M_IND;
  const float* __restrict__ b3 = Lu + (j0 + 3 * 16 + m) * M_IND;

  for (int k0 = 0; k0 < M_IND; k0 += 4) {
    v2f a;  a.x = arow[k0];        a.y = arow[k0 + 1];
    v2f q0; q0.x = b0[k0 + hi];    q0.y = b0[k0 + 2 + hi];
    v2f q1; q1.x = b1[k0 + hi];    q1.y = b1[k0 + 2 + hi];
    v2f q2; q2.x = b2[k0 + hi];    q2.y = b2[k0 + 2 + hi];
    v2f q3; q3.x = b3[k0 + hi];    q3.y = b3[k0 + 2 + hi];
    acc0 = __builtin_amdgcn_wmma_f32_16x16x4_f32(false, a, false, q0, (short)0, acc0, false, false);
    acc1 = __builtin_amdgcn_wmma_f32_16x16x4_f32(false, a, false, q1, (short)0, acc1, false, false);
    acc2 = __builtin_amdgcn_wmma_f32_16x16x4_f32(false, a, false, q2, (short)0, acc2, false, false);
    acc3 = __builtin_amdgcn_wmma_f32_16x16x4_f32(false, a, false, q3, (short)0, acc3, false, false);
  }

  float* __restrict__ outb = Su + (i0 + hi * 8) * M_IND + j0 + m;
#pragma unroll
  for (int r = 0; r < 8; ++r) {
    outb[r * M_IND + 0 * 16] = acc0[r];
    outb[r * M_IND + 1 * 16] = acc1[r];
    outb[r * M_IND + 2 * 16] = acc2[r];
    outb[r * M_IND + 3 * 16] = acc3[r];
  }
}

// ---------------------------------------------------------------------------
// Kernel 3: fused per-point pipeline. One wave32 per query point.
//   Z staged in LDS via async-to-LDS loads (ASYNCcnt) -> register-resident
//   distances -> top-16 via VALU argmin + butterfly -> gather 16x16 patches ->
//   register Gaussian elimination (lane-per-row, mirrored lanes) -> mean/std.
// ---------------------------------------------------------------------------
__global__ __launch_bounds__(256) void vnngp_kernel(
    const float* __restrict__ X, const float* __restrict__ Z,
    const float* __restrict__ Kzz, const float* __restrict__ Su,
    const float* __restrict__ mu, float* __restrict__ out) {
  __shared__ float zsh[M_IND * 3];          // 12 KB
  __shared__ int   idxS[NWAVES][K_NN];
  __shared__ float dvalS[NWAVES][K_NN];

  const int lane = threadIdx.x & 31;
  const int wave = threadIdx.x >> 5;

  // --- async stage Z (12 KB) into LDS: 768 x b128 chunks, ASYNCcnt-tracked ---
  {
    const unsigned ldsbase = (unsigned)(uintptr_t)&zsh[0];  // low 32 bits = LDS offset
    for (int t = threadIdx.x; t < 768; t += 256) {
      const unsigned ldsoff = ldsbase + (unsigned)t * 16u;
      const float* gp = Z + t * 4;
      asm volatile("global_load_async_to_lds_b128 %0, %1, off"
                   :: "v"(ldsoff), "v"(gp) : "memory");
    }
    asm volatile("s_wait_asynccnt 0" ::: "memory");
  }
  __syncthreads();

  const int n = blockIdx.x * NWAVES + wave;
  const float px = X[n * 3 + 0];
  const float py = X[n * 3 + 1];
  const float pz = X[n * 3 + 2];

  // all 1024 squared distances for this point, kept in registers (32/lane)
  float dloc[32];
#pragma unroll
  for (int c = 0; c < 32; ++c) {
    const int j = c * 32 + lane;
    const float dx = px - zsh[j * 3 + 0];
    const float dy = py - zsh[j * 3 + 1];
    const float dz = pz - zsh[j * 3 + 2];
    dloc[c] = fmaxf(dx * dx + dy * dy + dz * dz, 0.0f);
  }

  // top-16 smallest: VALU local argmin over masked registers + wave butterfly
  unsigned used = 0;
  for (int t = 0; t < K_NN; ++t) {
    float m = 3.0e38f;
    int bc = 0;
#pragma unroll
    for (int c = 0; c < 32; ++c) {
      const bool ok = !((used >> c) & 1u) && (dloc[c] < m);
      m = ok ? dloc[c] : m;
      bc = ok ? c : bc;
    }
    int bj = bc * 32 + lane;
    for (int off = 16; off; off >>= 1) {
      const float om = __shfl_xor(m, off);
      const int   oj = __shfl_xor(bj, off);
      if (om < m || (om == m && oj < bj)) { m = om; bj = oj; }
    }
    if ((bj & 31) == lane) used |= 1u << (bj >> 5);   // owner retires its entry
    if (lane == 0) { idxS[wave][t] = bj; dvalS[wave][t] = m; }
  }

  // lane r (and mirror lane r+16) owns row r of the 16x16 local system
  const int r = lane & 15;
  const int ir = idxS[wave][r];
  float bv = expf(-0.5f * dvalS[wave][r]);  // lKxz[r]
  const float lmu_r = mu[ir];

  float a[K_NN];   // lKzz + extra JITTER on diag (ref adds JITTER*eyeK again)
  float d[K_NN];   // lS - lKzz   (cov = 1 + W^T d W)
#pragma unroll
  for (int c = 0; c < K_NN; ++c) {
    const int ic = idxS[wave][c];
    const float kz = Kzz[ir * M_IND + ic];
    const float su = Su[ir * M_IND + ic];
    a[c] = kz + ((r == c) ? JITTER : 0.0f);
    d[c] = su - kz;
  }

  // forward elimination (SPD, no pivoting); pivot row broadcast via shfl
#pragma unroll
  for (int k = 0; k < K_NN; ++k) {
    const float pk = __shfl(a[k], k);
    const float pb = __shfl(bv, k);
    const float f = (r > k) ? (a[k] / pk) : 0.0f;
#pragma unroll
    for (int c = k + 1; c < K_NN; ++c) {
      const float pc = __shfl(a[c], k);
      a[c] -= f * pc;
    }
    bv -= f * pb;
  }

  // back substitution: lane r ends with x = W[r]
  float x = 0.0f;
#pragma unroll
  for (int k = K_NN - 1; k >= 0; --k) {
    const float num = __shfl(bv, k);
    const float den = __shfl(a[k], k);
    const float xk = num / den;
    if (r == k) x = xk;
    bv -= ((r < k) ? a[k] : 0.0f) * xk;
  }

  // mean = W . lmu ;  q = W^T (lS - lKzz) W
  float tr = 0.0f;
#pragma unroll
  for (int c = 0; c < K_NN; ++c) tr += d[c] * __shfl(x, c);

  float mpart = (lane < 16) ? x * lmu_r : 0.0f;
  float qpart = (lane < 16) ? x * tr : 0.0f;
  for (int off = 16; off; off >>= 1) {
    mpart += __shfl_xor(mpart, off);
    qpart += __shfl_xor(qpart, off);
  }

  if (lane == 0) {
    out[n] = mpart;
    const float cov = 1.0f + qpart;
    out[N_PTS + n] = sqrtf(fmaxf(cov, 0.05f));
  }
}

// ---------------------------------------------------------------------------
extern "C" void kernel_launch(void* const* d_in, const int* in_sizes, int n_in,
                              void* d_out, int out_size, void* d_ws, size_t ws_size,
                              hipStream_t stream) {
  const float* X      = (const float*)d_in[0];   // 65536 x 3
  const float* Z      = (const float*)d_in[1];   // 1024 x 3
  const float* Lu_raw = (const float*)d_in[2];   // 1024 x 1024
  const float* mu     = (const float*)d_in[3];   // 1024
  float* out = (float*)d_out;                    // 2 x 65536

  float* Lu  = (float*)d_ws;                     // 4 MB
  float* Kzz = Lu + M_IND * M_IND;               // 4 MB
  float* Su  = Kzz + M_IND * M_IND;              // 4 MB

  prep_kernel<<<(M_IND * M_IND) / 256, 256, 0, stream>>>(Lu_raw, Z, Lu, Kzz);
  su_gemm_wmma<<<dim3(M_IND / 64, (M_IND / 16) / NWAVES), 256, 0, stream>>>(Lu, Su);
  vnngp_kernel<<<N_PTS / NWAVES, 256, 0, stream>>>(X, Z, Kzz, Su, mu, out);
}